// AttentionLSTM_35794257445587
// MI455X (gfx1250) — compile-verified
//
#include <hip/hip_runtime.h>
#include <math.h>

typedef __attribute__((ext_vector_type(2))) float v2f;
typedef __attribute__((ext_vector_type(4))) float v4f;
typedef __attribute__((ext_vector_type(8))) float v8f;

constexpr int kB = 8;
constexpr int kS = 256;
constexpr int kD = 256;
constexpr int kN = 256;

// ---------------------------------------------------------------------------
// Hardware transcendentals (gfx1250): V_TANH_F32 is a native TRANS32 op.
// Prefer the clang builtin (gets proper TRANS-hazard scheduling); fall back
// to inline asm with a trailing v_nop to satisfy the 1-op TRANS hazard rule.
// ---------------------------------------------------------------------------
__device__ __forceinline__ float hw_tanh(float x) {
#if __has_builtin(__builtin_amdgcn_tanhf)
  return __builtin_amdgcn_tanhf(x);
#else
  float y;
  asm volatile("v_tanh_f32 %0, %1\n\tv_nop" : "=v"(y) : "v"(x));
  return y;
#endif
}

__device__ __forceinline__ float hw_rcp(float x) {
#if __has_builtin(__builtin_amdgcn_rcpf)
  return __builtin_amdgcn_rcpf(x);
#else
  return 1.0f / x;
#endif
}

// ---------------------------------------------------------------------------
// Kernel 1: Px = X @ Wx_w^T + Wx_b ; Pxh = X @ Wxhat_w^T + Wxhat_b
// One 16x16 output tile per wave, f32 WMMA 16x16x4, K = 256 (64 WMMAs/tile).
// A-lane layout: lane = m + 16*h, holds A[m][k+2h], A[m][k+2h+1].
// B-lane layout: lane = n + 16*h, holds B[k+2h][n], B[k+2h+1][n]
//   where B[kk][n] = W[n0+n][kk]  (W is [N,D] row-major -> contiguous pair).
// ---------------------------------------------------------------------------
__global__ __launch_bounds__(256) void proj_kernel(
    const float* __restrict__ X,      // [B*S, D]
    const float* __restrict__ Wx_w,   // [N, D]
    const float* __restrict__ Wx_b,   // [N]
    const float* __restrict__ Wxh_w,  // [N, D]
    const float* __restrict__ Wxh_b,  // [N]
    float* __restrict__ Px,           // [B*S, N]
    float* __restrict__ Pxh)          // [B*S, N]
{
  const int lane = threadIdx.x & 31;
  const int wave = threadIdx.x >> 5;
  const int g = blockIdx.x * 8 + wave;

  const int ntiles_n = kN / 16;                 // 16
  const int ntiles_m = (kB * kS) / 16;          // 128
  const int tiles_per_w = ntiles_m * ntiles_n;  // 2048

  const int w   = g / tiles_per_w;              // 0 -> Wx, 1 -> Wxhat
  const int rem = g % tiles_per_w;
  const int m0  = (rem / ntiles_n) * 16;
  const int n0  = (rem % ntiles_n) * 16;

  const float* Wmat = w ? Wxh_w : Wx_w;
  const float* bias = w ? Wxh_b : Wx_b;
  float*       Out  = w ? Pxh   : Px;

  const int m = lane & 15;   // also the column index n for B/C/D operands
  const int h = lane >> 4;

  const float* arow = X    + (size_t)(m0 + m) * kD;
  const float* brow = Wmat + (size_t)(n0 + m) * kD;

  v8f acc = {};
  #pragma unroll 4
  for (int k = 0; k < kD; k += 4) {
    v2f a = *(const v2f*)(arow + k + 2 * h);
    v2f b = *(const v2f*)(brow + k + 2 * h);
    acc = __builtin_amdgcn_wmma_f32_16x16x4_f32(
        /*neg_a=*/false, a, /*neg_b=*/false, b,
        /*c_mod=*/(short)0, acc, /*reuse_a=*/false, /*reuse_b=*/false);
  }

  const float bv = bias[n0 + m];
  #pragma unroll
  for (int r = 0; r < 8; ++r) {
    Out[(size_t)(m0 + r + 8 * h) * kN + n0 + m] = acc[r] + bv;
  }
}

// ---------------------------------------------------------------------------
// Kernel 2: A[b,i,j] = sigmoid( sum_n att_w[n]*tanh(Px[b,i,n]+Pxh[b,j,n]) + att_b )
// One 16x16 (i,j) tile per 256-thread block; Px/Pxh rows staged in LDS.
// Row pitch padded to 260 floats: stride%64 == 4 -> conflict-free across j,
// and 260*4 bytes keeps every row 16B-aligned for b128 LDS loads.
// Dominant stage: B*S*S*N = 134M hardware v_tanh_f32 evaluations.
// ---------------------------------------------------------------------------
__global__ __launch_bounds__(256) void attn_kernel(
    const float* __restrict__ Px,     // [B*S, N]
    const float* __restrict__ Pxh,    // [B*S, N]
    const float* __restrict__ att_w,  // [N]
    const float* __restrict__ att_b,  // [1]
    float* __restrict__ A)            // [B, S, S]
{
  constexpr int PITCH = kN + 4;       // 260
  __shared__ float sWx[16][PITCH];
  __shared__ float sWh[16][PITCH];
  __shared__ float sAw[kN];

  const int tid = threadIdx.x;
  const int b  = blockIdx.x >> 8;
  const int i0 = ((blockIdx.x >> 4) & 15) * 16;
  const int j0 = (blockIdx.x & 15) * 16;

  // Stage 16 rows x 256 floats of Px and Pxh (1024 float4 each; 4 per thread).
  #pragma unroll
  for (int r = 0; r < 4; ++r) {
    const int idx = r * 256 + tid;    // 0..1023
    const int row = idx >> 6;         // 0..15
    const int c4  = (idx & 63) << 2;  // 0..252, step 4
    v4f vx = *(const v4f*)(Px  + (size_t)(b * kS + i0 + row) * kN + c4);
    v4f vh = *(const v4f*)(Pxh + (size_t)(b * kS + j0 + row) * kN + c4);
    *(v4f*)&sWx[row][c4] = vx;
    *(v4f*)&sWh[row][c4] = vh;
  }
  sAw[tid] = att_w[tid];              // tid < 256 == kN
  __syncthreads();

  const int i = tid >> 4;
  const int j = tid & 15;
  const float* rx = &sWx[i][0];
  const float* rh = &sWh[j][0];

  float sum = 0.0f;
  #pragma unroll 2
  for (int n = 0; n < kN; n += 4) {
    v4f x  = *(const v4f*)(rx + n);
    v4f hh = *(const v4f*)(rh + n);
    v4f wv = *(const v4f*)(&sAw[n]);
    sum += wv.x * hw_tanh(x.x + hh.x);
    sum += wv.y * hw_tanh(x.y + hh.y);
    sum += wv.z * hw_tanh(x.z + hh.z);
    sum += wv.w * hw_tanh(x.w + hh.w);
  }
  const float z = sum + att_b[0];
  const float a = hw_rcp(1.0f + __expf(-z));
  A[(size_t)(b * kS + i0 + i) * kS + j0 + j] = a;
}

// ---------------------------------------------------------------------------
// Kernel 3: out[b] = A[b] @ X[b]   (per-batch 256x256x256 GEMM, f32 WMMA)
// ---------------------------------------------------------------------------
__global__ __launch_bounds__(256) void out_kernel(
    const float* __restrict__ A,  // [B, S, S]
    const float* __restrict__ X,  // [B, S, D]
    float* __restrict__ O)        // [B, S, D]
{
  const int lane = threadIdx.x & 31;
  const int wave = threadIdx.x >> 5;
  const int g = blockIdx.x * 8 + wave;

  const int b   = g >> 8;                 // 16*16 tiles per batch
  const int rem = g & 255;
  const int i0  = (rem >> 4) * 16;
  const int n0  = (rem & 15) * 16;

  const float* Ab = A + (size_t)b * kS * kS;
  const float* Xb = X + (size_t)b * kS * kD;
  float*       Ob = O + (size_t)b * kS * kD;

  const int m = lane & 15;
  const int h = lane >> 4;
  const float* arow = Ab + (size_t)(i0 + m) * kS;

  v8f acc = {};
  #pragma unroll 4
  for (int k = 0; k < kS; k += 4) {
    v2f a = *(const v2f*)(arow + k + 2 * h);
    v2f bv;
    bv.x = Xb[(size_t)(k + 2 * h)     * kD + n0 + m];
    bv.y = Xb[(size_t)(k + 2 * h + 1) * kD + n0 + m];
    acc = __builtin_amdgcn_wmma_f32_16x16x4_f32(
        false, a, false, bv, (short)0, acc, false, false);
  }

  #pragma unroll
  for (int r = 0; r < 8; ++r)
    Ob[(size_t)(i0 + r + 8 * h) * kD + n0 + m] = acc[r];
}

// ---------------------------------------------------------------------------
extern "C" void kernel_launch(void* const* d_in, const int* in_sizes, int n_in,
                              void* d_out, int out_size, void* d_ws, size_t ws_size,
                              hipStream_t stream) {
  const float* X     = (const float*)d_in[0];
  const float* Wx_w  = (const float*)d_in[1];
  const float* Wx_b  = (const float*)d_in[2];
  const float* Wxh_w = (const float*)d_in[3];
  const float* Wxh_b = (const float*)d_in[4];
  const float* att_w = (const float*)d_in[5];
  const float* att_b = (const float*)d_in[6];
  float* out = (float*)d_out;

  // Workspace layout (6 MB total):
  //   Px  [B*S*N] f32 : 2 MB
  //   Pxh [B*S*N] f32 : 2 MB
  //   A   [B*S*S] f32 : 2 MB
  float* Px  = (float*)d_ws;
  float* Pxh = Px  + (size_t)kB * kS * kN;
  float* Aw  = Pxh + (size_t)kB * kS * kN;

  // 4096 tiles / 8 waves per block
  proj_kernel<<<512, 256, 0, stream>>>(X, Wx_w, Wx_b, Wxh_w, Wxh_b, Px, Pxh);
  // B * (S/16) * (S/16) = 2048 blocks
  attn_kernel<<<2048, 256, 0, stream>>>(Px, Pxh, att_w, att_b, Aw);
  // 2048 tiles / 8 waves per block
  out_kernel<<<256, 256, 0, stream>>>(Aw, X, out);
}